// InterfaceTokenFoam_83794811945123
// MI455X (gfx1250) — compile-verified
//
#include <hip/hip_runtime.h>
#include <math.h>

// Problem constants
#define V_VOCAB 50257
#define D_DIM   128
#define T_SEQ   256
#define NSTEPS  5

// ws layout (float indices)
#define OFF_PARAMS 0      // [0..26] interaction, [27..29] step, [30..32] a_w, [33] sens, [34] mdb, [35..37] target_sim
#define OFF_MEM    64     // memories: 3*3*128 = 1152 floats
#define OFF_BPACK  1216   // packed WMMA B fragments: 64 row-pairs * 16 cols * 2 = 2048 floats
#define OFF_LOGITS 3264   // 50257 floats
#define OFF_MAXU   53524  // 1 uint (order-mapped running max)
#define OFF_Z      53525  // 1 float (softmax partition)
#define OFF_PARTZ  53536  // 128 floats
#define OFF_PARTW  53664  // 128 floats

typedef __attribute__((ext_vector_type(2))) float v2f;
typedef __attribute__((ext_vector_type(8))) float v8f;

__device__ __forceinline__ float wsum32(float v) {
#pragma unroll
  for (int m = 16; m >= 1; m >>= 1) v += __shfl_xor(v, m, 32);
  return v;
}

// ---------------------------------------------------------------------------
// One-time per-launch setup: interaction softmax, clipped step, sigmoid(a_w),
// |sens|, mdb, target_sim cached in ws; memories zeroed.
// ---------------------------------------------------------------------------
__global__ void init_kernel(const float* __restrict__ tension,
                            const float* __restrict__ temperature,
                            const float* __restrict__ target_sim,
                            const float* __restrict__ step_size,
                            const float* __restrict__ anchor_logit,
                            const float* __restrict__ mdb,
                            const float* __restrict__ nsens,
                            float* __restrict__ ws) {
  const int tid = threadIdx.x;
  if (tid < 9) {
    int k = tid / 3, n = tid % 3;
    float temp = fmaxf(fabsf(temperature[k]), 0.01f);
    float e0 = -tension[(k * 3 + n) * 3 + 0] / temp;
    float e1 = -tension[(k * 3 + n) * 3 + 1] / temp;
    float e2 = -tension[(k * 3 + n) * 3 + 2] / temp;
    float mx = fmaxf(e0, fmaxf(e1, e2));
    float z0 = expf(e0 - mx), z1 = expf(e1 - mx), z2 = expf(e2 - mx);
    float inv = 1.f / (z0 + z1 + z2);
    ws[k * 9 + n * 3 + 0] = z0 * inv;
    ws[k * 9 + n * 3 + 1] = z1 * inv;
    ws[k * 9 + n * 3 + 2] = z2 * inv;
  }
  if (tid < 3) {
    ws[27 + tid] = fminf(fmaxf(fabsf(step_size[tid]), 0.001f), 0.5f);
    ws[30 + tid] = 1.f / (1.f + expf(-anchor_logit[tid]));
    ws[35 + tid] = target_sim[tid];
  }
  if (tid == 0) { ws[33] = fabsf(nsens[0]); ws[34] = mdb[0]; }
  for (int i = tid; i < 1152; i += blockDim.x) ws[OFF_MEM + i] = 0.f;
}

// ---------------------------------------------------------------------------
// Per-token sequential dynamics: novelty/decay, state0, 5 equilibration
// iterations, memory update, 9x9 Gram -> Jacobi eigenvalues -> S, and packing
// of the rank-9 factor into WMMA B-fragment layout (scaled by 1/3).
// One workgroup (256 threads = 8 wave32), wave-per-row dot products.
// ---------------------------------------------------------------------------
__global__ __launch_bounds__(256) void dyn_kernel(const int* __restrict__ tokens,
                                                  const float* __restrict__ E,
                                                  const float* __restrict__ bases,
                                                  float* __restrict__ ws,
                                                  float* __restrict__ out,
                                                  int t) {
  __shared__ float s_x[128];
  __shared__ float s_mm[384];
  __shared__ float s_xwm[384];
  __shared__ float s_anchn[384];
  __shared__ float s_state[1152];
  __shared__ float s_expr[1152];
  __shared__ float s_exprn[1152];
  __shared__ float s_forces[1152];
  __shared__ float s_sc[8];
  __shared__ float s_decay[4];
  __shared__ float s_anorm[4];
  __shared__ float s_norm[12];
  __shared__ float s_cos[27];
  __shared__ float s_dnorm[27];
  __shared__ float s_wnm[27];
  __shared__ float s_asim[9];
  __shared__ float s_adn[9];
  __shared__ float s_awc[9];
  __shared__ float s_G[81];

  const int tid = threadIdx.x;
  const int wid = tid >> 5, lane = tid & 31;
  float* P = ws + OFF_PARAMS;
  float* mem = ws + OFF_MEM;

  const int tok = tokens[t];
  if (tid < 128) s_x[tid] = E[(size_t)tok * 128 + tid];
  for (int i = tid; i < 384; i += 256) {
    int k = i >> 7, d = i & 127;
    s_mm[i] = (mem[(k * 3 + 0) * 128 + d] + mem[(k * 3 + 1) * 128 + d] +
               mem[(k * 3 + 2) * 128 + d]) * (1.0f / 3.0f);
  }
  __syncthreads();

  // norms & dots for novelty: task0 = ||x||, 1..3 = ||mm_k||, 4..6 = mm_k . x
  for (int r = wid; r < 7; r += 8) {
    float p = 0.f;
#pragma unroll
    for (int jj = 0; jj < 4; ++jj) {
      int idx = jj * 32 + lane;
      if (r == 0)      { float v = s_x[idx]; p += v * v; }
      else if (r < 4)  { float v = s_mm[(r - 1) * 128 + idx]; p += v * v; }
      else             { p += s_mm[(r - 4) * 128 + idx] * s_x[idx]; }
    }
    p = wsum32(p);
    if (lane == 0) s_sc[r] = (r < 4) ? sqrtf(p) : p;
  }
  __syncthreads();

  if (tid < 3) {
    float xn = s_sc[0] + 1e-10f;
    float mn = s_sc[1 + tid] + 1e-10f;
    float cosv = s_sc[4 + tid] / (xn * mn);
    float nov = (mn > 1e-8f) ? (1.f - cosv) : 1.f;
    s_decay[tid] = 1.f / (1.f + expf(-(P[34] - P[33] * nov)));
  }
  __syncthreads();

  for (int i = tid; i < 384; i += 256) {
    int k = i >> 7;
    s_xwm[i] = s_x[i & 127] + s_decay[k] * s_mm[i];
  }
  __syncthreads();
  for (int r = wid; r < 3; r += 8) {
    float p = 0.f;
#pragma unroll
    for (int jj = 0; jj < 4; ++jj) { float v = s_xwm[r * 128 + jj * 32 + lane]; p += v * v; }
    p = wsum32(p);
    if (lane == 0) s_anorm[r] = sqrtf(p);
  }
  __syncthreads();
  for (int i = tid; i < 384; i += 256) {
    int k = i >> 7;
    s_anchn[i] = s_xwm[i] / (s_anorm[k] + 1e-10f);
  }
  // state0[kn][d] = sum_e xwm[k][e] * bases[kn][e][d]   (coalesced over d)
  for (int i = tid; i < 1152; i += 256) {
    int kn = i >> 7, d = i & 127, k = kn / 3;
    const float* B = bases + (size_t)kn * 16384;
    float acc = 0.f;
    for (int e = 0; e < 128; ++e) acc += s_xwm[k * 128 + e] * B[e * 128 + d];
    s_state[i] = acc;
  }
  __syncthreads();

  for (int it = 0; it < NSTEPS + 1; ++it) {
    // expr[kn][e] = bases[kn][e][:] . state[kn]   (wave-per-row dot)
    for (int r = wid; r < 1152; r += 8) {
      int kn = r >> 7, e = r & 127;
      const float* Brow = bases + (size_t)kn * 16384 + (size_t)e * 128;
      float p = 0.f;
#pragma unroll
      for (int jj = 0; jj < 4; ++jj) { int d = jj * 32 + lane; p += Brow[d] * s_state[kn * 128 + d]; }
      p = wsum32(p);
      if (lane == 0) s_expr[r] = p;
    }
    __syncthreads();
    for (int r = wid; r < 9; r += 8) {
      float p = 0.f;
#pragma unroll
      for (int jj = 0; jj < 4; ++jj) { float v = s_expr[r * 128 + jj * 32 + lane]; p += v * v; }
      p = wsum32(p);
      if (lane == 0) s_norm[r] = sqrtf(p) + 1e-10f;
    }
    __syncthreads();
    for (int i = tid; i < 1152; i += 256) s_exprn[i] = s_expr[i] / s_norm[i >> 7];
    __syncthreads();
    if (it == NSTEPS) break;

    // cos/diff-norm pairs (27) and anchor sims/diff-norms (9)
    for (int r = wid; r < 36; r += 8) {
      float pc = 0.f, pd = 0.f;
      if (r < 27) {
        int k = r / 9, n = (r / 3) % 3, m = r % 3;
        int a = (k * 3 + n) * 128, b = (k * 3 + m) * 128;
#pragma unroll
        for (int jj = 0; jj < 4; ++jj) {
          int idx = jj * 32 + lane;
          pc += s_exprn[a + idx] * s_exprn[b + idx];
          float dv = s_expr[a + idx] - s_expr[b + idx];
          pd += dv * dv;
        }
      } else {
        int kn = r - 27, k = kn / 3;
        int a = kn * 128, b = k * 128;
#pragma unroll
        for (int jj = 0; jj < 4; ++jj) {
          int idx = jj * 32 + lane;
          pc += s_exprn[a + idx] * s_anchn[b + idx];
          float dv = s_expr[a + idx] - s_xwm[b + idx];
          pd += dv * dv;
        }
      }
      pc = wsum32(pc); pd = wsum32(pd);
      if (lane == 0) {
        if (r < 27) { s_cos[r] = pc; s_dnorm[r] = sqrtf(pd); }
        else        { s_asim[r - 27] = pc; s_adn[r - 27] = sqrtf(pd); }
      }
    }
    __syncthreads();
    if (tid < 27) {
      int k = tid / 9, n = (tid / 3) % 3, m = tid % 3;
      float fm = (s_cos[tid] - P[35 + k]) * ((n == m) ? 0.f : P[k * 9 + n * 3 + m]);
      s_wnm[tid] = fm / (s_dnorm[tid] + 1e-10f);
    } else if (tid >= 32 && tid < 41) {
      int kn = tid - 32, k = kn / 3;
      float afm = (s_asim[kn] - P[35 + k]) * P[30 + k];
      s_awc[kn] = afm / (s_adn[kn] + 1e-10f);
    }
    __syncthreads();
    for (int i = tid; i < 1152; i += 256) {
      int kn = i >> 7, e = i & 127, k = kn / 3, n = kn % 3;
      float ev = s_expr[i];
      float f = 0.f;
#pragma unroll
      for (int m = 0; m < 3; ++m)
        f += s_wnm[k * 9 + n * 3 + m] * (ev - s_expr[(k * 3 + m) * 128 + e]);
      f += s_awc[kn] * (ev - s_xwm[k * 128 + e]);
      s_forces[i] = f;
    }
    __syncthreads();
    // f_local[kn][d] = sum_e forces[kn][e] * bases[kn][e][d]; state update
    for (int i = tid; i < 1152; i += 256) {
      int kn = i >> 7, d = i & 127, k = kn / 3;
      const float* B = bases + (size_t)kn * 16384;
      float acc = 0.f;
      for (int e = 0; e < 128; ++e) acc += s_forces[kn * 128 + e] * B[e * 128 + d];
      s_state[i] += P[27 + k] * acc;
    }
    __syncthreads();
  }

  // memories = decay*mem + (1-decay)*state
  for (int i = tid; i < 1152; i += 256) {
    int k = i / 384;
    float dk = s_decay[k];
    mem[i] = dk * mem[i] + (1.f - dk) * s_state[i];
  }
  // Gram matrix G[a][b] = exprn_a . exprn_b / 9 (== nonzero spectrum of rho)
  for (int r = wid; r < 81; r += 8) {
    int a = r / 9, b = r % 9;
    float p = 0.f;
#pragma unroll
    for (int jj = 0; jj < 4; ++jj) {
      int idx = jj * 32 + lane;
      p += s_exprn[a * 128 + idx] * s_exprn[b * 128 + idx];
    }
    p = wsum32(p);
    if (lane == 0) s_G[r] = p * (1.0f / 9.0f);
  }
  __syncthreads();
  // Pack WMMA B fragments: B[e][col] = exprn[col][e] / 3 (cols 9..15 zero).
  // Fragment layout for V_WMMA_F32_16X16X4_F32 B (4x16 f32, 2 VGPR):
  // VGPR j, lane half h, col n  ->  K-row = 2h + j  of each K-step.
  float* Bp = ws + OFF_BPACK;
  for (int i = tid; i < 2048; i += 256) {
    int cpair = i & 1, col = (i >> 1) & 15, ppair = i >> 5;
    int e = ppair * 2 + cpair;
    Bp[i] = (col < 9) ? s_exprn[col * 128 + e] * (1.0f / 3.0f) : 0.f;
  }
  if (tid == 0) {
    // 9x9 cyclic Jacobi eigenvalues (exact nonzero spectrum of rank-9 rho)
    float A[81];
    for (int i = 0; i < 81; ++i) A[i] = s_G[i];
    for (int sweep = 0; sweep < 12; ++sweep) {
      for (int p = 0; p < 9; ++p) {
        for (int q = p + 1; q < 9; ++q) {
          float apq = A[p * 9 + q];
          if (fabsf(apq) < 1e-20f) continue;
          float tau = (A[q * 9 + q] - A[p * 9 + p]) / (2.f * apq);
          float tt = ((tau >= 0.f) ? 1.f : -1.f) / (fabsf(tau) + sqrtf(1.f + tau * tau));
          float cc = 1.f / sqrtf(1.f + tt * tt);
          float ssn = tt * cc;
          for (int i2 = 0; i2 < 9; ++i2) {
            float aip = A[i2 * 9 + p], aiq = A[i2 * 9 + q];
            A[i2 * 9 + p] = cc * aip - ssn * aiq;
            A[i2 * 9 + q] = ssn * aip + cc * aiq;
          }
          for (int i2 = 0; i2 < 9; ++i2) {
            float api = A[p * 9 + i2], aqi = A[q * 9 + i2];
            A[p * 9 + i2] = cc * api - ssn * aqi;
            A[q * 9 + i2] = ssn * api + cc * aqi;
          }
        }
      }
    }
    float tot = 119.f * 1e-12f;  // the 119 clamped zero-eigenvalues
    float lam[9];
    for (int i = 0; i < 9; ++i) { lam[i] = fmaxf(A[i * 9 + i], 1e-12f); tot += lam[i]; }
    float S = 0.f;
    for (int i = 0; i < 9; ++i) { float qv = lam[i] / tot; S -= qv * fmaxf(logf(qv), -100.f); }
    float q0 = 1e-12f / tot;
    S -= 119.f * (q0 * fmaxf(logf(q0), -100.f));
    out[(size_t)T_SEQ * V_VOCAB + t] = S;
    ((unsigned*)(ws + OFF_MAXU))[0] = 0u;  // order-mapped -inf
  }
}

// ---------------------------------------------------------------------------
// Fused quadratic-form logits via f32 WMMA:
// logits[v] = sum_a (E[v,:] . exprn_a / 3)^2  using V_WMMA_F32_16X16X4_F32.
// One wave32 per 16-row tile of E; 32 WMMAs cover K=128.
// Also maintains the deterministic global max via order-mapped atomicMax.
// ---------------------------------------------------------------------------
__global__ __launch_bounds__(256) void gemm_logits_kernel(const float* __restrict__ E,
                                                          float* __restrict__ ws) {
  const float* Bpack = ws + OFF_BPACK;
  float* logits = ws + OFF_LOGITS;
  unsigned* maxslot = (unsigned*)(ws + OFF_MAXU);
  const int wid = threadIdx.x >> 5, lane = threadIdx.x & 31;
  const int tile = blockIdx.x * 8 + wid;
  const int ntiles = (V_VOCAB + 15) / 16;  // 3142
  if (tile >= ntiles) return;
  const int r0 = tile * 16;
  const int half = lane >> 4, j = lane & 15;
  const int arow = r0 + j;
  const int crow = (arow < V_VOCAB) ? arow : (V_VOCAB - 1);
  const float* Erow = E + (size_t)crow * 128;
  __builtin_prefetch(Erow, 0, 1);  // global_prefetch_b8 (L2-resident E)

  v8f c = {0.f, 0.f, 0.f, 0.f, 0.f, 0.f, 0.f, 0.f};
#pragma unroll
  for (int kk = 0; kk < 32; ++kk) {
    // A (16x4 f32): lane half h holds K = {2h, 2h+1} of this K-step
    const float2 av = *(const float2*)(Erow + kk * 4 + half * 2);
    // B (4x16 f32): packed so VGPR j holds K-row 2h+j, column = lane&15
    const float2 bv = *(const float2*)(Bpack + ((size_t)((kk * 2 + half) * 16 + j)) * 2);
    v2f A; A.x = av.x; A.y = av.y;
    v2f B; B.x = bv.x; B.y = bv.y;
    c = __builtin_amdgcn_wmma_f32_16x16x4_f32(false, A, false, B, (short)0, c,
                                              false, false);
  }
  // C layout: c[i] at lane -> row r0 + 8*half + i, col = lane&15.
  // Sum of squares over the 16 columns (cols 9..15 are exactly zero).
#pragma unroll
  for (int i = 0; i < 8; ++i) {
    float s = c[i] * c[i];
    s += __shfl_xor(s, 1, 32);
    s += __shfl_xor(s, 2, 32);
    s += __shfl_xor(s, 4, 32);
    s += __shfl_xor(s, 8, 32);
    int orow = r0 + half * 8 + i;
    if (j == i && orow < V_VOCAB) {
      logits[orow] = s;
      unsigned u = __float_as_uint(s);
      u = (u & 0x80000000u) ? ~u : (u | 0x80000000u);  // order-preserving map
      atomicMax(maxslot, u);                            // order-independent
    }
  }
}

// ---------------------------------------------------------------------------
// Softmax statistics: per-block partial Z = sum e^(l-max), W = sum (l-max)e^(l-max)
// ---------------------------------------------------------------------------
__global__ __launch_bounds__(256) void sum_kernel(float* __restrict__ ws) {
  const float* logits = ws + OFF_LOGITS;
  const unsigned mu = ((const unsigned*)(ws + OFF_MAXU))[0];
  const unsigned bits = (mu & 0x80000000u) ? (mu & 0x7FFFFFFFu) : ~mu;
  const float gmax = __uint_as_float(bits);
  const int tid = threadIdx.x;
  float z = 0.f, w = 0.f;
  for (int i = blockIdx.x * 256 + tid; i < V_VOCAB; i += 128 * 256) {
    float d = logits[i] - gmax;
    float e = expf(d);
    z += e; w += d * e;
  }
  __shared__ float sz[256], sw[256];
  sz[tid] = z; sw[tid] = w;
  __syncthreads();
  for (int s2 = 128; s2 > 0; s2 >>= 1) {
    if (tid < s2) { sz[tid] += sz[tid + s2]; sw[tid] += sw[tid + s2]; }
    __syncthreads();
  }
  if (tid == 0) { ws[OFF_PARTZ + blockIdx.x] = sz[0]; ws[OFF_PARTW + blockIdx.x] = sw[0]; }
}

__global__ __launch_bounds__(128) void final_kernel(float* __restrict__ ws,
                                                    float* __restrict__ out, int t) {
  __shared__ float sz[128], sw[128];
  const int tid = threadIdx.x;
  sz[tid] = ws[OFF_PARTZ + tid];
  sw[tid] = ws[OFF_PARTW + tid];
  __syncthreads();
  for (int s2 = 64; s2 > 0; s2 >>= 1) {
    if (tid < s2) { sz[tid] += sz[tid + s2]; sw[tid] += sw[tid + s2]; }
    __syncthreads();
  }
  if (tid == 0) {
    float Z = sz[0], W = sw[0];
    float H = logf(Z) - W / Z;  // H = logZ - E[l - max]
    size_t TV = (size_t)T_SEQ * V_VOCAB;
    float S = out[TV + t];
    out[TV + T_SEQ + t] = H;
    out[TV + 2 * T_SEQ + t] = H - S;
    ws[OFF_Z] = Z;
  }
}

__global__ __launch_bounds__(256) void prob_kernel(const float* __restrict__ ws,
                                                   float* __restrict__ out, int t) {
  const float* logits = ws + OFF_LOGITS;
  const unsigned mu = ((const unsigned*)(ws + OFF_MAXU))[0];
  const unsigned bits = (mu & 0x80000000u) ? (mu & 0x7FFFFFFFu) : ~mu;
  const float gmax = __uint_as_float(bits);
  const float invZ = 1.f / ws[OFF_Z];
  float* dst = out + (size_t)t * V_VOCAB;
  for (int i = blockIdx.x * 256 + threadIdx.x; i < V_VOCAB; i += (int)gridDim.x * 256)
    dst[i] = expf(logits[i] - gmax) * invZ;
}

extern "C" void kernel_launch(void* const* d_in, const int* in_sizes, int n_in,
                              void* d_out, int out_size, void* d_ws, size_t ws_size,
                              hipStream_t stream) {
  (void)in_sizes; (void)n_in; (void)out_size; (void)ws_size;
  const int*   tokens       = (const int*)d_in[0];
  const float* E            = (const float*)d_in[1];
  const float* bases        = (const float*)d_in[2];
  const float* tension      = (const float*)d_in[3];
  const float* temperature  = (const float*)d_in[4];
  const float* target_sim   = (const float*)d_in[5];
  const float* step_size    = (const float*)d_in[6];
  const float* anchor_logit = (const float*)d_in[7];
  const float* mdb          = (const float*)d_in[8];
  const float* nsens        = (const float*)d_in[9];
  float* ws  = (float*)d_ws;
  float* out = (float*)d_out;

  init_kernel<<<1, 128, 0, stream>>>(tension, temperature, target_sim, step_size,
                                     anchor_logit, mdb, nsens, ws);
  const int ntiles = (V_VOCAB + 15) / 16;          // 3142 row tiles
  const int gemm_blocks = (ntiles + 7) / 8;        // 8 wave32 per block
  for (int t = 0; t < T_SEQ; ++t) {
    dyn_kernel<<<1, 256, 0, stream>>>(tokens, E, bases, ws, out, t);
    gemm_logits_kernel<<<gemm_blocks, 256, 0, stream>>>(E, ws);
    sum_kernel<<<128, 256, 0, stream>>>(ws);
    final_kernel<<<1, 128, 0, stream>>>(ws, out, t);
    prob_kernel<<<197, 256, 0, stream>>>(ws, out, t);
  }
}